// DrugGCN_70600672411998
// MI455X (gfx1250) — compile-verified
//
#include <hip/hip_runtime.h>
#include <float.h>

#define FDIM 128

typedef __attribute__((ext_vector_type(2))) float v2f;
typedef __attribute__((ext_vector_type(8))) float v8f;

// ---------------------------------------------------------------- utilities

__global__ void zero_f32_kernel(float* __restrict__ p, long n) {
  long i = (long)blockIdx.x * blockDim.x + threadIdx.x;
  if (i < n) p[i] = 0.0f;
}

// deg[d] += 1 for every edge (incl. self-loops); deg buffer doubles as dinv.
__global__ void degree_kernel(const int* __restrict__ ei, float* __restrict__ deg,
                              int E, int N) {
  int e = blockIdx.x * blockDim.x + threadIdx.x;
  int EN = E + N;
  if (e >= EN) return;
  int d = (e < E) ? ei[E + e] : (e - E);
  atomicAdd(&deg[d], 1.0f);
}

__global__ void dinv_kernel(float* __restrict__ deg_dinv, int N) {
  int i = blockIdx.x * blockDim.x + threadIdx.x;
  if (i < N) deg_dinv[i] = rsqrtf(fmaxf(deg_dinv[i], 1.0f));
}

// ---------------------------------------------------------------- WMMA GEMM
// H[row0:row0+16, :] = X[row0:row0+16, 0:128] @ W[0:128, 0:128]
// One wave per 16x16 tile; 8 waves/block cover the 128 output columns.
// fp32 WMMA keeps full precision; GEMM is not the bottleneck here.
__global__ void gemm_wmma_kernel(const float* __restrict__ X,
                                 const float* __restrict__ W,
                                 float* __restrict__ H) {
  const int row0 = blockIdx.x * 16;
  const int col0 = (threadIdx.x >> 5) * 16;   // wave id * 16
  const int lane = threadIdx.x & 31;
  const int m    = lane & 15;                 // M for A, N for B/C
  const int kh   = lane >> 4;                 // half-wave: K offset 0 / 2

  v8f c = {};
  const float* xr = X + (size_t)(row0 + m) * FDIM;
  const float* wc = W + col0 + m;

  #pragma unroll 4
  for (int kk = 0; kk < FDIM / 4; ++kk) {
    const int k0 = kk * 4 + (kh ? 2 : 0);
    v2f a, b;
    a.x = xr[k0];
    a.y = xr[k0 + 1];
    b.x = wc[(size_t)k0 * FDIM];
    b.y = wc[(size_t)(k0 + 1) * FDIM];
    // D = A(16x4,f32) * B(4x16,f32) + C ; emits v_wmma_f32_16x16x4_f32
    c = __builtin_amdgcn_wmma_f32_16x16x4_f32(
        /*neg_a=*/false, a, /*neg_b=*/false, b,
        /*c_mod=*/(short)0, c, /*reuse_a=*/false, /*reuse_b=*/false);
  }

  // C/D layout: VGPR r -> (M = r, lanes 0-15) / (M = r+8, lanes 16-31), N = m
  float* hb = H + (size_t)(row0 + (kh ? 8 : 0)) * FDIM + col0 + m;
  #pragma unroll
  for (int r = 0; r < 8; ++r) hb[(size_t)r * FDIM] = c[r];
}

// Scalar tail for row counts not divisible by 16 (unused for N=100000).
__global__ void gemm_tail_kernel(const float* __restrict__ X,
                                 const float* __restrict__ W,
                                 float* __restrict__ H, int base) {
  const int row = base + blockIdx.x;
  const int f = threadIdx.x;
  const float* xr = X + (size_t)row * FDIM;
  float acc = 0.0f;
  #pragma unroll 8
  for (int k = 0; k < FDIM; ++k) acc = fmaf(xr[k], W[(size_t)k * FDIM + f], acc);
  H[(size_t)row * FDIM + f] = acc;
}

// ---------------------------------------------------------------- aggregation
// One wave per edge: G[dst] += H[src] * dinv[src]*dinv[dst]; 4 feats/lane.
__global__ void aggregate_kernel(const float* __restrict__ H,
                                 const int* __restrict__ ei,
                                 const float* __restrict__ dinv,
                                 float* __restrict__ G, int E, int N) {
  const int e = blockIdx.x * (blockDim.x >> 5) + (threadIdx.x >> 5);
  const int EN = E + N;
  if (e >= EN) return;
  const int lane = threadIdx.x & 31;
  int s, d;
  if (e < E) { s = ei[e]; d = ei[E + e]; }
  else       { s = e - E; d = s; }            // self-loop
  const float w = dinv[s] * dinv[d];
  const float4 hv = reinterpret_cast<const float4*>(H + (size_t)s * FDIM)[lane];
  float* gp = G + (size_t)d * FDIM + lane * 4;
  atomicAdd(gp + 0, hv.x * w);
  atomicAdd(gp + 1, hv.y * w);
  atomicAdd(gp + 2, hv.z * w);
  atomicAdd(gp + 3, hv.w * w);
}

__global__ void bias_relu_kernel(float* __restrict__ G, const float* __restrict__ b) {
  const size_t i = (size_t)blockIdx.x * FDIM + threadIdx.x;
  G[i] = fmaxf(G[i] + b[threadIdx.x], 0.0f);
}

// ---------------------------------------------------------------- pooling
__global__ void init_out_kernel(float* __restrict__ out, int ng) {
  int i = blockIdx.x * blockDim.x + threadIdx.x;
  if (i >= ng * 2 * FDIM) return;
  int f = i & (2 * FDIM - 1);
  out[i] = (f < FDIM) ? 0.0f : __int_as_float(0xFF7FFFFF); // 0 for sum, -FLT_MAX for max
}

__device__ __forceinline__ void atomic_max_f32(float* addr, float v) {
  if (v >= 0.0f) atomicMax((int*)addr, __float_as_int(v));
  else           atomicMin((unsigned int*)addr, (unsigned int)__float_as_int(v));
}

__global__ void pool_kernel(const float* __restrict__ G, const int* __restrict__ gi,
                            float* __restrict__ out, float* __restrict__ cnt) {
  const int node = blockIdx.x;
  const int f = threadIdx.x;
  const int g = gi[node];
  const float v = G[(size_t)node * FDIM + f];
  atomicAdd(&out[(size_t)g * 2 * FDIM + f], v);
  atomic_max_f32(&out[(size_t)g * 2 * FDIM + FDIM + f], v);
  if (f == 0) atomicAdd(&cnt[g], 1.0f);
}

__global__ void finalize_mean_kernel(float* __restrict__ out, const float* __restrict__ cnt) {
  const int g = blockIdx.x;
  out[(size_t)g * 2 * FDIM + threadIdx.x] /= fmaxf(cnt[g], 1.0f);
}

// ---------------------------------------------------------------- launch

extern "C" void kernel_launch(void* const* d_in, const int* in_sizes, int n_in,
                              void* d_out, int out_size, void* d_ws, size_t ws_size,
                              hipStream_t stream) {
  const float* x  = (const float*)d_in[0];
  const int*   ei = (const int*)d_in[1];
  const int*   gi = (const int*)d_in[2];
  const float* Wm[3] = {(const float*)d_in[4], (const float*)d_in[6], (const float*)d_in[8]};
  const float* Bv[3] = {(const float*)d_in[5], (const float*)d_in[7], (const float*)d_in[9]};
  float* out = (float*)d_out;

  const int  N  = in_sizes[0] / FDIM;
  const int  E  = in_sizes[1] / 2;
  const int  NG = out_size / (2 * FDIM);
  const long Nf = (long)N * FDIM;

  // workspace layout (16B-aligned blocks first for float4 loads)
  float* Hbuf = (float*)d_ws;      // [N,128] transform output
  float* Gbuf = Hbuf + Nf;         // [N,128] aggregation output / next input
  float* dinv = Gbuf + Nf;         // [N]     degree^-1/2 (deg accumulated in-place)
  float* cnt  = dinv + N;          // [NG]
  if (ws_size < (size_t)(2 * Nf + N + NG) * sizeof(float)) return;

  auto zero = [&](float* p, long n) {
    zero_f32_kernel<<<(unsigned)((n + 255) / 256), 256, 0, stream>>>(p, n);
  };

  // symmetric normalization
  zero(dinv, N);
  degree_kernel<<<(E + N + 255) / 256, 256, 0, stream>>>(ei, dinv, E, N);
  dinv_kernel<<<(N + 255) / 256, 256, 0, stream>>>(dinv, N);

  const int tiles = N / 16;
  const int rem   = N - tiles * 16;
  const float* cur = x;
  for (int l = 0; l < 3; ++l) {
    if (tiles > 0)
      gemm_wmma_kernel<<<tiles, 256, 0, stream>>>(cur, Wm[l], Hbuf);
    if (rem > 0)
      gemm_tail_kernel<<<rem, FDIM, 0, stream>>>(cur, Wm[l], Hbuf, tiles * 16);
    zero(Gbuf, Nf);
    aggregate_kernel<<<(E + N + 7) / 8, 256, 0, stream>>>(Hbuf, ei, dinv, Gbuf, E, N);
    bias_relu_kernel<<<N, FDIM, 0, stream>>>(Gbuf, Bv[l]);
    cur = Gbuf;
  }

  // global mean/max pooling
  init_out_kernel<<<(NG * 2 * FDIM + 255) / 256, 256, 0, stream>>>(out, NG);
  zero(cnt, NG);
  pool_kernel<<<N, FDIM, 0, stream>>>(Gbuf, gi, out, cnt);
  finalize_mean_kernel<<<NG, FDIM, 0, stream>>>(out, cnt);
}